// SS2D_56813827391776
// MI455X (gfx1250) — compile-verified
//
#include <hip/hip_runtime.h>
#include <hip/hip_bf16.h>
#include <math.h>

// ---------------------------------------------------------------------------
// SS2D (VMamba selective-scan block) for MI455X / gfx1250, fp32 end-to-end.
// B=8, H=W=64 (L=4096), C=Dn=192, N=8, R=12, K=4. BL = 32768 tokens.
// GEMMs: V_WMMA_F32_16X16X4_F32 (wave32 WMMA) with the weight panel DMA'd
// into LDS by the Tensor Data Mover (tensor_load_to_lds + s_wait_tensorcnt),
// using TDM hardware padding (2 DWORDs / 64 DWORDs) for a bank-conflict-free
// LDS layout. Scan is fused (no xs/dts materialization) + atomic merge.
// ---------------------------------------------------------------------------

typedef float        v2f __attribute__((ext_vector_type(2)));
typedef float        v8f __attribute__((ext_vector_type(8)));
typedef unsigned int v4u __attribute__((ext_vector_type(4)));
typedef int          v4i __attribute__((ext_vector_type(4)));
typedef int          v8i __attribute__((ext_vector_type(8)));

#define BL    32768      // B * L tokens
#define DN    192
#define LSEQ  4096
#define HW    64

// Padded LDS row stride: 192 data DWORDs + 2 pad DWORDs per 64 (TDM pad).
// element (row, col) -> lds[row*198 + col + 2*(col>>6)]
#define LDSSTR 198

#if defined(__AMDGCN__) && __has_builtin(__builtin_amdgcn_tensor_load_to_lds) && \
    __has_builtin(__builtin_amdgcn_s_wait_tensorcnt)
#define USE_TDM 1
#else
#define USE_TDM 0
#endif

// ---------------------------------------------------------------------------
// Generic WMMA f32 GEMM:  C(M,N) = A(M,Kd) @ W(N,Kd)^T    (Kd <= 192, Kd%4==0)
// block = 128 threads (4 waves); block tile = 64(M) x 64(N).
// Wave w -> 16 rows x 64 cols (4 v8f accumulators), K-loop step 4.
// W panel (64 x Kd) brought into LDS by one TDM 2D-tile load (wave 0 issues).
// ---------------------------------------------------------------------------
__device__ __forceinline__ void store_tile(float* __restrict__ C, v8f acc,
                                           int m0, int ncol, int ldc) {
#pragma unroll
  for (int v = 0; v < 8; ++v) {
    C[(size_t)(m0 + v) * ldc + ncol] = acc[v];
  }
}

__global__ __launch_bounds__(128) void gemm_wmma_f32(
    const float* __restrict__ A, const float* __restrict__ W,
    float* __restrict__ C, int M, int N, int Kd, int ldc) {
  __shared__ float lds[64 * LDSSTR];

  const int n0   = blockIdx.x * 64;
  const int mblk = blockIdx.y * 64;
  const int tid  = threadIdx.x;

#if USE_TDM
  // --- Tensor Data Mover: DMA W rows [n0, n0+64) x Kd into LDS -------------
  // D# group0: count=1 | lds_addr | global_addr(57b) | type=2.
  // D# group1: data_size=4B, pad_enable, pad_interval=64 DW, pad_amount=2 DW,
  //            tensor_dim0=Kd, tensor_dim1=N-n0 (OOB rows -> zero-fill),
  //            tile_dim0=Kd, tile_dim1=64, dim0_stride=Kd.
  if ((tid >> 5) == 0) {                     // wave 0 only (wave-uniform)
    unsigned lds_off = (unsigned)(size_t)(void*)&lds[0];     // LDS byte offset
    unsigned long long ga = (unsigned long long)(const void*)(W + (size_t)n0 * Kd);
    int nrem = N - n0;
    v4u g0 = { 1u,                                           // count=1 (valid)
               lds_off,
               (unsigned)(ga & 0xFFFFFFFFull),
               (unsigned)(((ga >> 32) & 0x01FFFFFFull) | 0x80000000u) }; // type=2
    v8i g1 = { (int)0x03520000u,                             // ds=4B|pad|int64|amt2
               (int)(((unsigned)Kd & 0xFFFFu) << 16),        // tensor_dim0 lo
               (int)((((unsigned)nrem & 0xFFFFu) << 16) | (((unsigned)Kd >> 16) & 0xFFFFu)),
               (int)((((unsigned)Kd & 0xFFFFu) << 16) | (((unsigned)nrem >> 16) & 0xFFFFu)),
               64,                                           // tile_dim1=64, tile_dim2=0
               Kd,                                           // tensor_dim0_stride lo
               0, 0 };
    v4i gz4 = { 0, 0, 0, 0 };
    v8i gz8 = { 0, 0, 0, 0, 0, 0, 0, 0 };
    __builtin_amdgcn_tensor_load_to_lds(g0, g1, gz4, gz4, gz8, 0);
    __builtin_amdgcn_s_wait_tensorcnt(0);
  }
#else
  // Fallback: cooperative load into the same padded layout (float2 stores;
  // row*198 parity is even so every even column is 8B-aligned).
  const int nq = 64 * (Kd >> 1);             // float2 count
  for (int q = tid; q < nq; q += 128) {
    int row = q / (Kd >> 1);
    int c2  = (q - row * (Kd >> 1)) << 1;
    int n   = n0 + row;
    v2f val = {0.f, 0.f};
    if (n < N) val = *(const v2f*)&W[(size_t)n * Kd + c2];
    *(v2f*)&lds[row * LDSSTR + c2 + ((c2 >> 6) << 1)] = val;
  }
#endif
  __syncthreads();

  const int wave = tid >> 5;
  const int lane = tid & 31;
  const int l16  = lane & 15;
  const int koff = (lane >> 4) << 1;         // 0 or 2
  const int m0   = mblk + wave * 16;

  v8f a0 = {}, a1 = {}, a2 = {}, a3 = {};

  const float* Ap = A + (size_t)(m0 + l16) * Kd + koff;
  __builtin_prefetch(Ap, 0, 3);              // global_prefetch_b8 of the A row

  for (int kk = 0; kk < Kd; kk += 4) {
    // padded-column base: col + 2*(col>>6); koff (0/2) never crosses a
    // 64-DWORD boundary for kk%4==0, so the adjust is uniform per kk.
    const float* Lq = lds + kk + ((kk >> 6) << 1) + koff;
    v2f a  = *(const v2f*)(Ap + kk);
    v2f b0 = *(const v2f*)(Lq + (l16      ) * LDSSTR);
    v2f b1 = *(const v2f*)(Lq + (l16 + 16 ) * LDSSTR);
    v2f b2 = *(const v2f*)(Lq + (l16 + 32 ) * LDSSTR);
    v2f b3 = *(const v2f*)(Lq + (l16 + 48 ) * LDSSTR);
    a0 = __builtin_amdgcn_wmma_f32_16x16x4_f32(false, a, false, b0, (short)0, a0, false, false);
    a1 = __builtin_amdgcn_wmma_f32_16x16x4_f32(false, a, false, b1, (short)0, a1, false, false);
    a2 = __builtin_amdgcn_wmma_f32_16x16x4_f32(false, a, false, b2, (short)0, a2, false, false);
    a3 = __builtin_amdgcn_wmma_f32_16x16x4_f32(false, a, false, b3, (short)0, a3, false, false);
  }

  // C/D layout: VGPR v -> row m0+v (lanes 0-15) / m0+v+8 (lanes 16-31),
  // col = n0 + jtile*16 + (lane & 15).
  const int mrow = m0 + ((lane >> 4) << 3);
  if (n0      + l16 < N) store_tile(C, a0, mrow, n0      + l16, ldc);
  if (n0 + 16 + l16 < N) store_tile(C, a1, mrow, n0 + 16 + l16, ldc);
  if (n0 + 32 + l16 < N) store_tile(C, a2, mrow, n0 + 32 + l16, ldc);
  if (n0 + 48 + l16 < N) store_tile(C, a3, mrow, n0 + 48 + l16, ldc);
}

// ---------------------------------------------------------------------------
// Depthwise 3x3 conv + SiLU.  Input xf = columns [0,192) of xz (row stride
// 384); output xc in (B,H,W,Dn) with Dn contiguous (coalesced).
// ---------------------------------------------------------------------------
__global__ __launch_bounds__(256) void conv3x3_silu(
    const float* __restrict__ xz, const float* __restrict__ cw,
    const float* __restrict__ cb, float* __restrict__ xc) {
  int idx = blockIdx.x * 256 + threadIdx.x;      // over BL*DN
  int d    = idx % DN;
  int rest = idx / DN;                           // b*L + h*W + w
  int w    = rest % HW;
  int h    = (rest / HW) % HW;
  int bofs = rest - h * HW - w;                  // b*L

  float acc = cb[d];
#pragma unroll
  for (int dy = 0; dy < 3; ++dy) {
    int hh = h + dy - 1;
    if ((unsigned)hh >= HW) continue;
#pragma unroll
    for (int dx = 0; dx < 3; ++dx) {
      int ww = w + dx - 1;
      if ((unsigned)ww >= HW) continue;
      acc += xz[(size_t)(bofs + hh * HW + ww) * 384 + d] * cw[d * 9 + dy * 3 + dx];
    }
  }
  float sg = 1.f / (1.f + __expf(-acc));
  xc[idx] = acc * sg;                            // SiLU
}

// ---------------------------------------------------------------------------
// Selective scan, one block per (b, direction-k), 192 threads = Dn lanes.
// h[8] states in registers; per step the 28 projected values (R=12, B=8, C=8)
// are broadcast through LDS; delta = softplus(P_R . dt_w[d] + bias) computed
// in-register; y contribution atomically accumulated at the SPATIAL location
// (this implements the flip / transpose-back merge of the 4 directions).
// ---------------------------------------------------------------------------
__global__ __launch_bounds__(192) void scan_dir(
    const float* __restrict__ xc, const float* __restrict__ P,
    const float* __restrict__ dtw_g, const float* __restrict__ dtb_g,
    const float* __restrict__ Alog, float* __restrict__ yacc) {
  const int bk = blockIdx.x;
  const int k  = bk & 3;
  const int b  = bk >> 2;
  const int d  = threadIdx.x;

  float dtw[12];
#pragma unroll
  for (int r = 0; r < 12; ++r) dtw[r] = dtw_g[(size_t)(k * DN + d) * 12 + r];
  const float bias = dtb_g[k * DN + d];

  float Aneg[8];
#pragma unroll
  for (int n = 0; n < 8; ++n) Aneg[n] = -__expf(Alog[(size_t)(k * DN + d) * 8 + n]);

  float h[8];
#pragma unroll
  for (int n = 0; n < 8; ++n) h[n] = 0.f;

  __shared__ float sh[28];
  const size_t base = (size_t)b * LSEQ;

  for (int t = 0; t < LSEQ; ++t) {
    int l;
    if (k == 0)      l = t;
    else if (k == 1) l = ((t & 63) << 6) | (t >> 6);          // wh order
    else if (k == 2) l = (LSEQ - 1) - t;                      // reversed hw
    else { int tp = (LSEQ - 1) - t; l = ((tp & 63) << 6) | (tp >> 6); }

    __syncthreads();
    if (d < 28) sh[d] = P[(base + l) * 112 + k * 28 + d];
    __syncthreads();

    float u  = xc[(base + l) * DN + d];
    float rd = bias;
#pragma unroll
    for (int r = 0; r < 12; ++r) rd += sh[r] * dtw[r];
    float delta = (rd > 20.f) ? rd : log1pf(__expf(rd));      // softplus
    float du = delta * u;

    float y = 0.f;
#pragma unroll
    for (int n = 0; n < 8; ++n) {
      h[n] = __expf(delta * Aneg[n]) * h[n] + du * sh[12 + n];
      y += h[n] * sh[20 + n];
    }
    atomicAdd(&yacc[(base + l) * DN + d], y);                 // global_atomic_add_f32
  }
}

// ---------------------------------------------------------------------------
// y = yacc + (sum_k Ds[k,:]) * xc ; LayerNorm over Dn ; * SiLU(z) -> yg.
// One wave32 per token (B*L = 32768 rows), shfl reductions.
// ---------------------------------------------------------------------------
__global__ __launch_bounds__(256) void ln_gate(
    const float* __restrict__ yacc, const float* __restrict__ xc,
    const float* __restrict__ xz, const float* __restrict__ Ds,
    const float* __restrict__ lnw, const float* __restrict__ lnb,
    float* __restrict__ yg) {
  const int row  = blockIdx.x * 8 + (threadIdx.x >> 5);
  const int lane = threadIdx.x & 31;

  float v[6];
  float s = 0.f, ss = 0.f;
#pragma unroll
  for (int i = 0; i < 6; ++i) {
    int d = lane + 32 * i;
    float dsum = Ds[d] + Ds[DN + d] + Ds[2 * DN + d] + Ds[3 * DN + d];
    float val  = yacc[(size_t)row * DN + d] + dsum * xc[(size_t)row * DN + d];
    v[i] = val; s += val; ss += val * val;
  }
#pragma unroll
  for (int off = 16; off > 0; off >>= 1) {
    s  += __shfl_down(s, off, 32);
    ss += __shfl_down(ss, off, 32);
  }
  s  = __shfl(s, 0, 32);
  ss = __shfl(ss, 0, 32);
  const float mu   = s * (1.f / DN);
  const float var  = ss * (1.f / DN) - mu * mu;
  const float rstd = rsqrtf(var + 1e-5f);

#pragma unroll
  for (int i = 0; i < 6; ++i) {
    int d = lane + 32 * i;
    float yn = (v[i] - mu) * rstd * lnw[d] + lnb[d];
    float zz = xz[(size_t)row * 384 + DN + d];                // z half of xz
    float sg = 1.f / (1.f + __expf(-zz));
    yg[(size_t)row * DN + d] = yn * (zz * sg);
  }
}

// ---------------------------------------------------------------------------
// Host-side orchestration (graph-capture safe: only async ops on `stream`).
// ---------------------------------------------------------------------------
extern "C" void kernel_launch(void* const* d_in, const int* in_sizes, int n_in,
                              void* d_out, int out_size, void* d_ws, size_t ws_size,
                              hipStream_t stream) {
  const float* x      = (const float*)d_in[0];   // (8,64,64,192)
  const float* inpw   = (const float*)d_in[1];   // (384,192)
  const float* convw  = (const float*)d_in[2];   // (192,1,3,3)
  const float* convb  = (const float*)d_in[3];   // (192)
  const float* xprojw = (const float*)d_in[4];   // (4,28,192) -> flat (112,192)
  const float* dtw    = (const float*)d_in[5];   // (4,192,12)
  const float* dtb    = (const float*)d_in[6];   // (4,192)
  const float* Alog   = (const float*)d_in[7];   // (4,192,8)
  const float* Ds     = (const float*)d_in[8];   // (4,192)
  const float* lnw    = (const float*)d_in[9];   // (192)
  const float* lnb    = (const float*)d_in[10];  // (192)
  const float* outw   = (const float*)d_in[11];  // (192,192)
  float* out = (float*)d_out;                    // (8,64,64,192)

  char* ws = (char*)d_ws;
  const size_t szXZ   = (size_t)BL * 384 * sizeof(float);  // 48 MB
  const size_t szDN   = (size_t)BL * DN  * sizeof(float);  // 24 MB
  const size_t szP    = (size_t)BL * 112 * sizeof(float);  // 14 MB
  float* xz   = (float*)(ws);
  float* xc   = (float*)(ws + szXZ);
  float* Pbuf = (float*)(ws + szXZ + szDN);
  float* yacc = (float*)(ws + szXZ + szDN + szP);
  float* yg   = (float*)(ws + szXZ + szDN + szP + szDN);

  (void)hipMemsetAsync(yacc, 0, szDN, stream);

  // 1) xz = x @ in_proj_w^T              (32768 x 192) @ (192 x 384)
  gemm_wmma_f32<<<dim3(6, BL / 64), 128, 0, stream>>>(x, inpw, xz, BL, 384, DN, 384);

  // 2) depthwise 3x3 conv + SiLU on xf half of xz
  conv3x3_silu<<<(BL * DN) / 256, 256, 0, stream>>>(xz, convw, convb, xc);

  // 3) P = xc @ x_proj_w_cat^T           (32768 x 192) @ (192 x 112)
  gemm_wmma_f32<<<dim3(2, BL / 64), 128, 0, stream>>>(xc, xprojw, Pbuf, BL, 112, DN, 112);

  // 4) 4-direction selective scan, atomic spatial merge
  scan_dir<<<8 * 4, DN, 0, stream>>>(xc, Pbuf, dtw, dtb, Alog, yacc);

  // 5) skip term + LayerNorm + SiLU(z) gate
  ln_gate<<<BL / 8, 256, 0, stream>>>(yacc, xc, xz, Ds, lnw, lnb, yg);

  // 6) out = yg @ out_proj_w^T           (32768 x 192) @ (192 x 192)
  gemm_wmma_f32<<<dim3(3, BL / 64), 128, 0, stream>>>(yg, outw, out, BL, DN, DN, DN);
}